// GCNGraph_87917980549524
// MI455X (gfx1250) — compile-verified
//
#include <hip/hip_runtime.h>
#include <cstddef>

#define BN_EPS_F 1e-5f
#define NEG_SLOPE_F 0.2f

typedef __attribute__((ext_vector_type(2))) float v2f;
typedef __attribute__((ext_vector_type(8))) float v8f;

__device__ __forceinline__ v8f wmma_f32(v2f a, v2f b, v8f c)
{
    return __builtin_amdgcn_wmma_f32_16x16x4_f32(false, a, false, b,
                                                 (short)0, c, false, false);
}

// A-fragment loader.  AMODE 0: row-major; 1: mask (A>0 || m==k) -> {0,1};
// 2: transposed (A[k][m]).
template <int AMODE>
__device__ __forceinline__ v2f load_afrag(const float* __restrict__ A, int lda,
                                          int mrow, int ka)
{
    v2f a;
    if (AMODE == 0) {
        a.x = A[(size_t)mrow * lda + ka];
        a.y = A[(size_t)mrow * lda + ka + 1];
    } else if (AMODE == 1) {
        float t0 = A[(size_t)mrow * lda + ka];
        float t1 = A[(size_t)mrow * lda + ka + 1];
        a.x = (t0 > 0.f || mrow == ka)     ? 1.f : 0.f;
        a.y = (t1 > 0.f || mrow == ka + 1) ? 1.f : 0.f;
    } else {
        a.x = A[(size_t)ka * lda + mrow];
        a.y = A[(size_t)(ka + 1) * lda + mrow];
    }
    return a;
}

__device__ __forceinline__ v2f load_bfrag(const float* __restrict__ Bp, int ldb,
                                          int ka, int off)
{
    const float* p = Bp + (size_t)ka * ldb + off;
    v2f b;
    b.x = p[0];
    b.y = p[ldb];
    return b;
}

// ---------------------------------------------------------------------------
// fp32 WMMA GEMM:  C(MxN) = op(A)(MxK) @ B(KxN) + epilogue
// Each wave computes a 32(M) x 64(N) strip: 2 A-fragments x 4 B-fragments =
// 8 independent WMMA accumulator chains per K-step; A/B fragments for the
// next K-step are preloaded (register double-buffering) so v_wmma does not
// wait on just-issued loads.  Block = 4 waves stacked in M (128x64 tile);
// grid = (N/64, M/128).  Requires M%128==0, N%64==0, K%4==0, K>=8.
// Epilogue: optional bias; optional BN+ReLU; optional zero-diagonal.
// ---------------------------------------------------------------------------
template <int AMODE>
__global__ void k_wmma_gemm(const float* __restrict__ A, int lda,
                            const float* __restrict__ B, int ldb,
                            float* __restrict__ C, int ldc,
                            int M, int N, int K,
                            const float* __restrict__ bias,
                            const float* __restrict__ bng,
                            const float* __restrict__ bnb,
                            const float* __restrict__ bnm,
                            const float* __restrict__ bnv,
                            int zero_diag)
{
    int lane = threadIdx.x & 31;
    int wave = threadIdx.x >> 5;
    int m0 = blockIdx.y * 128 + wave * 32;
    int n0 = blockIdx.x * 64;
    if (m0 >= M || n0 >= N) return;        // wave-uniform guard

    int r     = lane & 15;
    int khalf = (lane >> 4) << 1;          // 0 (lanes 0-15) / 2 (lanes 16-31)
    int mr0   = m0 + r;
    int mr1   = m0 + 16 + r;

    v8f acc[8];
    #pragma unroll
    for (int i = 0; i < 8; ++i) acc[i] = (v8f){0.f,0.f,0.f,0.f,0.f,0.f,0.f,0.f};

    const float* Bp = B + n0 + r;          // column base for this lane

    // prologue: fragments for k-step 0
    v2f a0 = load_afrag<AMODE>(A, lda, mr0, khalf);
    v2f a1 = load_afrag<AMODE>(A, lda, mr1, khalf);
    v2f b0 = load_bfrag(Bp, ldb, khalf, 0);
    v2f b1 = load_bfrag(Bp, ldb, khalf, 16);
    v2f b2 = load_bfrag(Bp, ldb, khalf, 32);
    v2f b3 = load_bfrag(Bp, ldb, khalf, 48);

    for (int k0 = 0; k0 + 4 < K; k0 += 4) {
        int kn = k0 + 4 + khalf;
        // preload next step's fragments (independent of the WMMAs below)
        v2f a0n = load_afrag<AMODE>(A, lda, mr0, kn);
        v2f a1n = load_afrag<AMODE>(A, lda, mr1, kn);
        v2f c0  = load_bfrag(Bp, ldb, kn, 0);
        v2f c1  = load_bfrag(Bp, ldb, kn, 16);
        v2f c2  = load_bfrag(Bp, ldb, kn, 32);
        v2f c3  = load_bfrag(Bp, ldb, kn, 48);

        acc[0] = wmma_f32(a0, b0, acc[0]);
        acc[4] = wmma_f32(a1, b0, acc[4]);
        acc[1] = wmma_f32(a0, b1, acc[1]);
        acc[5] = wmma_f32(a1, b1, acc[5]);
        acc[2] = wmma_f32(a0, b2, acc[2]);
        acc[6] = wmma_f32(a1, b2, acc[6]);
        acc[3] = wmma_f32(a0, b3, acc[3]);
        acc[7] = wmma_f32(a1, b3, acc[7]);

        a0 = a0n; a1 = a1n;
        b0 = c0; b1 = c1; b2 = c2; b3 = c3;
    }
    // epilogue k-step
    acc[0] = wmma_f32(a0, b0, acc[0]);
    acc[4] = wmma_f32(a1, b0, acc[4]);
    acc[1] = wmma_f32(a0, b1, acc[1]);
    acc[5] = wmma_f32(a1, b1, acc[5]);
    acc[2] = wmma_f32(a0, b2, acc[2]);
    acc[6] = wmma_f32(a1, b2, acc[6]);
    acc[3] = wmma_f32(a0, b3, acc[3]);
    acc[7] = wmma_f32(a1, b3, acc[7]);

    // store + epilogue
    #pragma unroll
    for (int mi = 0; mi < 2; ++mi) {
        int rbase = m0 + 16 * mi + ((lane >> 4) << 3);
        #pragma unroll
        for (int t = 0; t < 4; ++t) {
            int ncol = n0 + 16 * t + r;
            #pragma unroll
            for (int v = 0; v < 8; ++v) {
                int row = rbase + v;
                float val = acc[mi * 4 + t][v];
                if (bias) val += bias[ncol];
                if (bng) {
                    val = bng[ncol] * (val - bnm[ncol])
                              * rsqrtf(bnv[ncol] + BN_EPS_F) + bnb[ncol];
                    val = fmaxf(val, 0.f);
                }
                if (zero_diag && row == ncol) val = 0.f;
                C[(size_t)row * ldc + ncol] = val;
            }
        }
    }
}

// ---- zero a small vector ----
__global__ void k_zero(float* p, int n)
{
    int i = blockIdx.x * blockDim.x + threadIdx.x;
    if (i < n) p[i] = 0.f;
}

// ---- dinv[i] = rsqrt(1 + sum_j A[i][j])   (A symmetric; self-loop +1) ----
__global__ void k_dinv(const float* __restrict__ A, float* __restrict__ dinv, int n)
{
    __shared__ float red[256];
    int i = blockIdx.x, t = threadIdx.x;
    float s = 0.f;
    for (int j = t; j < n; j += 256) s += A[(size_t)i * n + j];
    red[t] = s; __syncthreads();
    for (int st = 128; st; st >>= 1) {
        if (t < st) red[t] += red[t + st];
        __syncthreads();
    }
    if (t == 0) dinv[i] = rsqrtf(red[0] + 1.f);
}

// ---- Ahat[i][j] = dinv[i] * (A[i][j] + (i==j)) * dinv[j] ----
__global__ void k_ahat(const float* __restrict__ A, const float* __restrict__ dinv,
                       float* __restrict__ Ahat, int n)
{
    size_t idx = (size_t)blockIdx.x * 256 + threadIdx.x;
    int i = (int)(idx / n), j = (int)(idx % n);
    float v = A[idx] + ((i == j) ? 1.f : 0.f);
    Ahat[idx] = dinv[i] * v * dinv[j];
}

// ---- masked column max: out[j][f] = max_{i in mask col j} X[i][f]
//      mask(i,j) = A[i][j]>0 || i==j ; A symmetric -> read row j. ----
__global__ void k_colmax(const float* __restrict__ A, const float* __restrict__ X,
                         float* __restrict__ out, int n)
{
    int j = blockIdx.x, f = threadIdx.x;     // 256 threads = feature dim
    const float* Arow = A + (size_t)j * n;
    float m = -INFINITY;
    for (int i = 0; i < n; ++i) {
        if (Arow[i] > 0.f || i == j)
            m = fmaxf(m, X[(size_t)i * 256 + f]);
    }
    out[(size_t)j * 256 + f] = m;
}

// ---- per-row dot: out[r] = dot(X[r,:256], w) + (bias ? *bias : 0) ----
__global__ void k_rowdot(const float* __restrict__ X, const float* __restrict__ w,
                         const float* __restrict__ bias, float* __restrict__ out, int n)
{
    int row  = blockIdx.x * 8 + (threadIdx.x >> 5);
    int lane = threadIdx.x & 31;
    if (row >= n) return;
    float s = 0.f;
    for (int f = lane; f < 256; f += 32) s += X[(size_t)row * 256 + f] * w[f];
    for (int off = 16; off; off >>= 1) s += __shfl_xor(s, off, 32);
    if (lane == 0) out[row] = s + (bias ? bias[0] : 0.f);
}

// ---- attention + column softmax; St[j][i] = S[i][j] (S transposed) ----
__global__ void k_attn(const float* __restrict__ A, const float* __restrict__ aj,
                       const float* __restrict__ aq, const float* __restrict__ attb,
                       float* __restrict__ St, int n)
{
    __shared__ float sm[2048];
    __shared__ float red[256];
    int j = blockIdx.x, t = threadIdx.x;
    float aqj = aq[j] + attb[0];
    float lmax = -INFINITY;
    for (int i = t; i < n; i += 256) {
        bool msk = (A[(size_t)j * n + i] > 0.f) || (i == j);  // A symmetric
        float v;
        if (msk) {
            v = aj[i] + aqj;
            v = (v > 0.f) ? v : NEG_SLOPE_F * v;
        } else {
            v = -INFINITY;
        }
        sm[i] = v;
        lmax = fmaxf(lmax, v);
    }
    red[t] = lmax; __syncthreads();
    for (int s = 128; s; s >>= 1) {
        if (t < s) red[t] = fmaxf(red[t], red[t + s]);
        __syncthreads();
    }
    float mx = red[0]; __syncthreads();
    float lsum = 0.f;
    for (int i = t; i < n; i += 256) {
        float e = expf(sm[i] - mx);
        sm[i] = e;
        lsum += e;
    }
    red[t] = lsum; __syncthreads();
    for (int s = 128; s; s >>= 1) {
        if (t < s) red[t] += red[t + s];
        __syncthreads();
    }
    float inv = 1.f / red[0];
    for (int i = t; i < n; i += 256) St[(size_t)j * n + i] = sm[i] * inv;
}

// ---- fit[i] = sigmoid( sum_j mask(i,j)*u[j] - deg(i)*v2[i] + v3[i] + le3b ) ----
__global__ void k_fit(const float* __restrict__ A, const float* __restrict__ u,
                      const float* __restrict__ v2, const float* __restrict__ v3,
                      const float* __restrict__ le3b, float* __restrict__ fit, int n)
{
    __shared__ float rs[256], rd[256];
    int i = blockIdx.x, t = threadIdx.x;
    float su = 0.f, sd = 0.f;
    for (int j = t; j < n; j += 256) {
        if (A[(size_t)i * n + j] > 0.f || j == i) { su += u[j]; sd += 1.f; }
    }
    rs[t] = su; rd[t] = sd; __syncthreads();
    for (int s = 128; s; s >>= 1) {
        if (t < s) { rs[t] += rs[t + s]; rd[t] += rd[t + s]; }
        __syncthreads();
    }
    if (t == 0) {
        float val = rs[0] - rd[0] * v2[i] + v3[i] + le3b[0];
        fit[i] = 1.f / (1.f + expf(-val));
    }
}

// ---- single-block bitonic top-k (n power of two <= 2048), descending,
//      ties broken by smaller index first (matches jax.lax.top_k) ----
__global__ void k_topk(const float* __restrict__ fit, float* __restrict__ fitk,
                       int* __restrict__ perm, int n, int k)
{
    __shared__ float v[2048];
    __shared__ int   id[2048];
    int t = threadIdx.x;                               // 1024 threads
    for (int i = t; i < n; i += 1024) { v[i] = fit[i]; id[i] = i; }
    __syncthreads();
    for (int size = 2; size <= n; size <<= 1) {
        for (int stride = size >> 1; stride > 0; stride >>= 1) {
            for (int q = t; q < (n >> 1); q += 1024) {
                int a = (q / stride) * (stride << 1) + (q % stride);
                int b = a + stride;
                bool desc  = ((a & size) == 0);
                bool aLess = (v[a] < v[b]) || (v[a] == v[b] && id[a] > id[b]);
                if (desc ? aLess : !aLess) {
                    float tv = v[a]; v[a] = v[b]; v[b] = tv;
                    int   ti = id[a]; id[a] = id[b]; id[b] = ti;
                }
            }
            __syncthreads();
        }
    }
    for (int c = t; c < k; c += 1024) { fitk[c] = v[c]; perm[c] = id[c]; }
}

// ---- Sp[i][c] = St[perm[c]][i]  (tiled gather-transpose) ----
__global__ void k_spgather(const float* __restrict__ St, const int* __restrict__ perm,
                           float* __restrict__ Sp, int n, int k)
{
    __shared__ float tile[16][17];
    int c0 = blockIdx.x * 16, i0 = blockIdx.y * 16;
    int tx = threadIdx.x, ty = threadIdx.y;
    tile[ty][tx] = St[(size_t)perm[c0 + ty] * n + i0 + tx];
    __syncthreads();
    Sp[(size_t)(i0 + ty) * k + c0 + tx] = tile[tx][ty];
}

// ---- x_out[c][f] = x_new[perm[c]][f] * fitk[c] ----
__global__ void k_xout(const float* __restrict__ xnew, const int* __restrict__ perm,
                       const float* __restrict__ fitk, float* __restrict__ xout, int k)
{
    int idx = blockIdx.x * 256 + threadIdx.x;
    int c = idx >> 8, f = idx & 255;
    xout[idx] = xnew[(size_t)perm[c] * 256 + f] * fitk[c];
}

// ---- readout[f] += mean_c x_out[c][f]  (single block, 256 threads) ----
__global__ void k_readout(const float* __restrict__ xout, float* __restrict__ ro, int k)
{
    int f = threadIdx.x;
    float s = 0.f;
    for (int c = 0; c < k; ++c) s += xout[(size_t)c * 256 + f];
    ro[f] += s / (float)k;
}

// ---- out[o] = dot(readout, W[:,o]) + b[o]  (single block, 128 threads) ----
__global__ void k_final(const float* __restrict__ ro, const float* __restrict__ W,
                        const float* __restrict__ b, float* __restrict__ out)
{
    int o = threadIdx.x;
    float s = 0.f;
    for (int f = 0; f < 256; ++f) s += ro[f] * W[(size_t)f * 128 + o];
    out[o] = s + b[o];
}

// ---------------------------------------------------------------------------
static void gemm(hipStream_t s, const float* A, int lda, const float* B, int ldb,
                 float* C, int ldc, int M, int N, int K,
                 const float* bias, const float* g, const float* be,
                 const float* mu, const float* var, int a_mode, int zdiag)
{
    dim3 grid(N / 64, M / 128), block(128);
    if (a_mode == 0)
        k_wmma_gemm<0><<<grid, block, 0, s>>>(A, lda, B, ldb, C, ldc, M, N, K,
                                              bias, g, be, mu, var, zdiag);
    else if (a_mode == 1)
        k_wmma_gemm<1><<<grid, block, 0, s>>>(A, lda, B, ldb, C, ldc, M, N, K,
                                              bias, g, be, mu, var, zdiag);
    else
        k_wmma_gemm<2><<<grid, block, 0, s>>>(A, lda, B, ldb, C, ldc, M, N, K,
                                              bias, g, be, mu, var, zdiag);
}

// Input layout (setup_inputs() insertion order, nested dicts/lists flattened
// in insertion/index order):
//  0:x  1:A  2..21:p1{Ws0,Ws1,Ws2,bs0..2,gamma0..2,beta0..2,mean0..2,var0..2,Wl,bl}
//  22..41:p2(same)  42..51:pasap{lin_W,lin_b,att_q,att_j,att_b,le1_W,le1_b,le2_W,le3_W,le3_b}
//  52:lin_W  53:lin_b
enum { IN_X = 0, IN_A = 1, P1 = 2, P2 = 22, PA = 42, IN_LINW = 52, IN_LINB = 53 };

extern "C" void kernel_launch(void* const* d_in, const int* in_sizes, int n_in,
                              void* d_out, int out_size, void* d_ws, size_t ws_size,
                              hipStream_t stream)
{
    (void)in_sizes; (void)n_in; (void)out_size; (void)ws_size;
    auto F = [&](int i) { return (const float*)d_in[i]; };

    // ---- workspace bump allocator (~66 MB total) ----
    char* wp = (char*)d_ws;
    auto alloc = [&](size_t bytes) -> void* {
        void* p = wp;
        wp += (bytes + 255) & ~(size_t)255;
        return p;
    };
    float* Ahat = (float*)alloc(2048ull * 2048 * 4);
    float* St   = (float*)alloc(2048ull * 2048 * 4);
    float* Sp   = (float*)alloc(2048ull * 1024 * 4);
    float* MfSp = (float*)alloc(2048ull * 1024 * 4);
    float* Ab1  = (float*)alloc(1024ull * 1024 * 4);
    float* Ab2  = (float*)alloc(512ull * 512 * 4);
    float* XA   = (float*)alloc(2048ull * 256 * 4);
    float* XB   = (float*)alloc(2048ull * 256 * 4);
    float* XW   = (float*)alloc(2048ull * 256 * 4);
    float* xqp  = (float*)alloc(2048ull * 256 * 4);
    float* xq   = (float*)alloc(2048ull * 256 * 4);
    float* xnew = (float*)alloc(2048ull * 256 * 4);
    float* dinv = (float*)alloc(2048 * 4);
    float* aj   = (float*)alloc(2048 * 4);
    float* aq   = (float*)alloc(2048 * 4);
    float* uu   = (float*)alloc(2048 * 4);
    float* vv2  = (float*)alloc(2048 * 4);
    float* vv3  = (float*)alloc(2048 * 4);
    float* fit  = (float*)alloc(2048 * 4);
    float* fitk = (float*)alloc(2048 * 4);
    int*   perm = (int*)alloc(2048 * 4);
    float* ro   = (float*)alloc(256 * 4);

    k_zero<<<1, 256, 0, stream>>>(ro, 256);

    const float* Xin  = F(IN_X);
    int          ldin = 128;
    const float* Acur = F(IN_A);
    float*       Anext[2] = { Ab1, Ab2 };

    int Nn = 2048;
    for (int r = 0; r < 3; ++r) {
        int base = (r == 0) ? P1 : P2;
        int kk = Nn / 2;

        // ---- normalized adjacency ----
        k_dinv<<<Nn, 256, 0, stream>>>(Acur, dinv, Nn);
        k_ahat<<<(unsigned)(((size_t)Nn * Nn) / 256), 256, 0, stream>>>(Acur, dinv, Ahat, Nn);

        // ---- GCN base: 3x (conv + BN + ReLU) + final conv ----
        const float* cur = Xin;
        int ldc_ = ldin;
        for (int l = 0; l < 4; ++l) {
            const float* Wl_ = (l < 3) ? F(base + l)     : F(base + 18);
            const float* bl_ = (l < 3) ? F(base + 3 + l) : F(base + 19);
            int Kd = (r == 0 && l == 0) ? 128 : 256;
            gemm(stream, cur, ldc_, Wl_, 256, XW, 256, Nn, 256, Kd,
                 nullptr, nullptr, nullptr, nullptr, nullptr, 0, 0);
            float* out = (cur == XA) ? XB : XA;
            if (l < 3)
                gemm(stream, Ahat, Nn, XW, 256, out, 256, Nn, 256, Nn,
                     bl_, F(base + 6 + l), F(base + 9 + l),
                     F(base + 12 + l), F(base + 15 + l), 0, 0);
            else
                gemm(stream, Ahat, Nn, XW, 256, out, 256, Nn, 256, Nn,
                     bl_, nullptr, nullptr, nullptr, nullptr, 0, 0);
            cur = out;
            ldc_ = 256;
        }
        const float* Xh = cur;

        // ---- ASAP pooling ----
        k_colmax<<<Nn, 256, 0, stream>>>(Acur, Xh, xqp, Nn);
        gemm(stream, xqp, 256, F(PA + 0), 256, xq, 256, Nn, 256, 256,
             F(PA + 1), nullptr, nullptr, nullptr, nullptr, 0, 0);
        k_rowdot<<<Nn / 8, 256, 0, stream>>>(Xh, F(PA + 3), nullptr, aj, Nn);
        k_rowdot<<<Nn / 8, 256, 0, stream>>>(xq, F(PA + 2), nullptr, aq, Nn);
        k_attn<<<Nn, 256, 0, stream>>>(Acur, aj, aq, F(PA + 4), St, Nn);
        gemm(stream, St, Nn, Xh, 256, xnew, 256, Nn, 256, Nn,
             nullptr, nullptr, nullptr, nullptr, nullptr, 0, 0);
        k_rowdot<<<Nn / 8, 256, 0, stream>>>(xnew, F(PA + 5), F(PA + 6), uu, Nn);
        k_rowdot<<<Nn / 8, 256, 0, stream>>>(xnew, F(PA + 7), nullptr, vv2, Nn);
        k_rowdot<<<Nn / 8, 256, 0, stream>>>(xnew, F(PA + 8), nullptr, vv3, Nn);
        k_fit<<<Nn, 256, 0, stream>>>(Acur, uu, vv2, vv3, F(PA + 9), fit, Nn);
        k_topk<<<1, 1024, 0, stream>>>(fit, fitk, perm, Nn, kk);

        float* xout = (Xh == XA) ? XB : XA;
        k_xout<<<kk, 256, 0, stream>>>(xnew, perm, fitk, xout, kk);
        k_readout<<<1, 256, 0, stream>>>(xout, ro, kk);

        if (r < 2) {   // A_new unused after last round
            dim3 tg(kk / 16, Nn / 16), tb(16, 16);
            k_spgather<<<tg, tb, 0, stream>>>(St, perm, Sp, Nn, kk);
            // MfSp = Mf @ Sp  (Mf built on the fly from Acur)
            gemm(stream, Acur, Nn, Sp, kk, MfSp, kk, Nn, kk, Nn,
                 nullptr, nullptr, nullptr, nullptr, nullptr, 1, 0);
            // A_new = Sp^T @ MfSp, zero diagonal
            gemm(stream, Sp, kk, MfSp, kk, Anext[r], kk, kk, kk, Nn,
                 nullptr, nullptr, nullptr, nullptr, nullptr, 2, 1);
            Acur = Anext[r];
        }
        Xin = xout;
        ldin = 256;
        Nn = kk;
    }

    k_final<<<1, 128, 0, stream>>>(ro, F(IN_LINW), F(IN_LINB), (float*)d_out);
}